// ContinuousWaveletTransform_81003083202981
// MI455X (gfx1250) — compile-verified
//
#include <hip/hip_runtime.h>

typedef __attribute__((ext_vector_type(2))) float v2f;
typedef __attribute__((ext_vector_type(8))) float v8f;

#define NSC   32
#define LSIG  4096
#define KMAX  2048
#define XPAD  (KMAX + LSIG)   // 6144 padded samples per batch
#define TAPS  16              // fp32 Morlet envelope is exactly 0 past t=15

// wl per scale, matching np.linspace(1,64,32) + int() truncation + clamps + evening.
// constexpr -> usable both at compile time (schedule) and on-device (bank builder).
constexpr int wavelen(int i) {
    double s = (i == 31) ? 64.0 : 1.0 + (63.0 / 31.0) * (double)i;
    int wl = (int)(64.0 * s);
    if (wl > KMAX) wl = KMAX;
    if (wl < 8)    wl = 8;
    if (wl & 1)    wl += 1;
    return wl;
}

// Compile-time schedule of active 4-wide K-blocks for a 16-scale tile.
// Rows iterated in increasing-klo order; kb_next watermark dedupes blocks
// (each WMMA consumes all 16 rows of A, so visiting a block once is exact).
struct Sched { int n; int kb[96]; };

constexpr Sched build_sched(int sb) {
    Sched s{0, {}};
    int kb_next = 0;
    for (int i = 15; i >= 0; --i) {
        int klo = KMAX - wavelen(sb + i);
        int b0  = klo & ~3;
        int b1  = (klo + TAPS - 1) & ~3;
        int kb  = (b0 > kb_next) ? b0 : kb_next;
        for (; kb <= b1; kb += 4) s.kb[s.n++] = kb;
        if (b1 + 4 > kb_next) kb_next = b1 + 4;
    }
    return s;
}

constexpr Sched SCHED_LO = build_sched(0);    // scales 0..15: disjoint 16-tap windows
constexpr Sched SCHED_HI = build_sched(16);   // scales 16..31: all wl==2048 -> blocks 0,4,8,12
static_assert(SCHED_LO.n == 72, "lo schedule size changed");
static_assert(SCHED_HI.n == 4,  "hi schedule size changed");

// Build left-aligned kernel bank: bank[ch][k], ch 0..31 = re(scale), 32..63 = im(scale).
__global__ void cwt_build_bank(float* __restrict__ bank) {
    const int ch    = blockIdx.x;        // 0..63
    const int scale = ch & 31;
    const int isIm  = ch >> 5;
    const int klo   = KMAX - wavelen(scale);
    const double PI = 3.14159265358979323846;
    for (int k = threadIdx.x; k < KMAX; k += blockDim.x) {
        float v = 0.0f;
        int t = k - klo;
        if (t >= 0 && t < TAPS) {
            double td  = (double)t;
            double env = exp(-0.5 * td * td);
            double ph  = (2.0 * PI / 6.0) * td;
            v = (float)(isIm ? env * sin(ph) : env * cos(ph));
        }
        bank[ch * KMAX + k] = v;
    }
}

// Zero-pad signal on the left by KMAX: xp[b][j] = signal[b][j - KMAX] or 0.
__global__ void cwt_build_xpad(const float* __restrict__ sig, float* __restrict__ xp) {
    int idx = blockIdx.x * blockDim.x + threadIdx.x;
    if (idx >= 4 * XPAD) return;
    int b = idx / XPAD, j = idx - b * XPAD;
    int src = j - KMAX;
    xp[idx] = (src >= 0 && src < LSIG) ? sig[b * LSIG + src] : 0.0f;
}

// One 16x16x4 fp32 WMMA step at compile-time K-block offset kb.
__device__ __forceinline__ v8f wstep(v8f acc, const float* __restrict__ arow,
                                     const float* __restrict__ brow, int kb) {
    v2f a = *(const v2f*)(arow + kb);   // even index -> 8B aligned
    v2f b;
    b.x = brow[kb];
    b.y = brow[kb + 1];
    return __builtin_amdgcn_wmma_f32_16x16x4_f32(false, a, false, b,
                                                 (short)0, acc, false, false);
}

// One wave computes a 16(ch) x 16(n) output tile. Fully unrolled constant-offset
// K schedule: 2 global_load_b64 + 1 v_wmma per active block, no runtime bookkeeping.
__global__ __launch_bounds__(32)
void cwt_wmma(const float* __restrict__ bank, const float* __restrict__ xp,
              float* __restrict__ out) {
    const int n0   = blockIdx.x * 16;      // output column tile
    const int mt   = blockIdx.y;           // 0..3 channel tile (16 channels)
    const int b    = blockIdx.z;           // batch
    const int c0   = mt * 16;              // first channel of tile
    const int lane = threadIdx.x;
    const int lm   = lane & 15;
    const int kk   = (lane >> 4) * 2;      // K sub-pair per fp32 WMMA A/B layout

    // A fragment base: row (c0+lm) of bank, K pair at kb+kk
    const float* __restrict__ arow = bank + (size_t)(c0 + lm) * KMAX + kk;
    // B fragment base: B[k, n] = xp[b][n0 + n + k]
    const float* __restrict__ brow = xp + (size_t)b * XPAD + n0 + lm + kk;

    v8f acc = {};

    if ((mt & 1) == 0) {
        #pragma unroll
        for (int t = 0; t < SCHED_LO.n; ++t)
            acc = wstep(acc, arow, brow, SCHED_LO.kb[t]);
    } else {
        #pragma unroll
        for (int t = 0; t < SCHED_HI.n; ++t)
            acc = wstep(acc, arow, brow, SCHED_HI.kb[t]);
    }

    // Scatter into interleaved complex64 output: out[b][scale][n].{re,im}
    const int hi  = lane >> 4;
    const int col = n0 + lm;
    #pragma unroll
    for (int r = 0; r < 8; ++r) {
        int ch    = c0 + r + 8 * hi;                 // C/D layout: M = r + 8*(lane>=16)
        int scale = ch & 31;
        int part  = ch >> 5;                         // 0 = re, 1 = im
        out[(((size_t)b * NSC + scale) * LSIG + col) * 2 + part] = acc[r];
    }
}

extern "C" void kernel_launch(void* const* d_in, const int* in_sizes, int n_in,
                              void* d_out, int out_size, void* d_ws, size_t ws_size,
                              hipStream_t stream) {
    const float* sig = (const float*)d_in[0];      // (4, 4096) fp32
    float* out = (float*)d_out;                    // (4, 32, 4096) complex64 interleaved

    float* bank = (float*)d_ws;                    // 64 * 2048 floats = 512 KB
    float* xp   = bank + 64 * KMAX;                // 4 * 6144 floats  =  96 KB

    cwt_build_bank<<<64, 256, 0, stream>>>(bank);
    cwt_build_xpad<<<(4 * XPAD + 255) / 256, 256, 0, stream>>>(sig, xp);

    dim3 grid(LSIG / 16, 4, 4);                    // 256 n-tiles x 4 ch-tiles x 4 batches
    cwt_wmma<<<grid, 32, 0, stream>>>(bank, xp, out);
}